// ChebNetwork_71691594105494
// MI455X (gfx1250) — compile-verified
//
#include <hip/hip_runtime.h>

typedef float v2f __attribute__((ext_vector_type(2)));
typedef float v4f __attribute__((ext_vector_type(4)));
typedef float v8f __attribute__((ext_vector_type(8)));

#if __has_builtin(__builtin_amdgcn_global_load_async_to_lds_b32)
#define HAVE_ASYNC_LDS 1
#else
#define HAVE_ASYNC_LDS 0
#endif

typedef __attribute__((address_space(1))) int g_i32;  // global int
typedef __attribute__((address_space(3))) int l_i32;  // LDS int

static __device__ __forceinline__ void atomic_add_f32(float* p, float v) {
  __hip_atomic_fetch_add(p, v, __ATOMIC_RELAXED, __HIP_MEMORY_SCOPE_AGENT);
}

__global__ __launch_bounds__(256) void k_zero(float* __restrict__ p, long n) {
  long i = (long)blockIdx.x * blockDim.x + threadIdx.x;
  if (i < n) p[i] = 0.0f;
}

__global__ __launch_bounds__(256) void k_negcopy(float* __restrict__ d,
                                                 const float* __restrict__ s, long n) {
  long i = (long)blockIdx.x * blockDim.x + threadIdx.x;
  if (i < n) d[i] = -s[i];
}

__global__ __launch_bounds__(256) void k_deg(const long long* __restrict__ ei,
                                             const float* __restrict__ ew,
                                             float* __restrict__ deg, int E) {
  int e = blockIdx.x * blockDim.x + threadIdx.x;
  if (e >= E) return;
  int r = (int)ei[e];
  int c = (int)ei[E + e];
  float w = (r == c) ? 0.0f : ew[e];
  atomic_add_f32(deg + r, w);
}

__global__ __launch_bounds__(256) void k_dinv(float* __restrict__ deg, int N) {
  int i = blockIdx.x * blockDim.x + threadIdx.x;
  if (i >= N) return;
  float d = deg[i];
  deg[i] = (d > 0.0f) ? rsqrtf(d) : 0.0f;
}

__global__ __launch_bounds__(256) void k_norm(const long long* __restrict__ ei,
                                              const float* __restrict__ ew,
                                              const float* __restrict__ dinv,
                                              float* __restrict__ nrm, int E) {
  int e = blockIdx.x * blockDim.x + threadIdx.x;
  if (e >= E) return;
  int r = (int)ei[e];
  int c = (int)ei[E + e];
  float w = (r == c) ? 0.0f : ew[e];
  nrm[e] = -dinv[r] * w * dinv[c];
}

// Scatter: y[col] += scale*norm[e]*x[row]. 16 threads/edge (D=64) or 8 (D=32),
// float4 coalesced gather, 4 hardware global_atomic_add_f32 per thread.
template <int DIN>
__global__ __launch_bounds__(256) void k_spmm(const long long* __restrict__ ei,
                                              const float* __restrict__ nrm,
                                              const float* __restrict__ x,
                                              float* __restrict__ y,
                                              int E, float scale) {
  constexpr int CH = DIN / 4;
  long tid = (long)blockIdx.x * blockDim.x + threadIdx.x;
  if (tid >= (long)E * CH) return;
  int e = (int)(tid / CH);
  int c = (int)(tid - (long)e * CH);
  float s = scale * nrm[e];
  int r = (int)ei[e];
  int col = (int)ei[E + e];
  if (s != 0.0f) {
    const v4f xv = *(const v4f*)(x + (size_t)r * DIN + 4 * c);
    float* yp = y + (size_t)col * DIN + 4 * c;
    atomic_add_f32(yp + 0, s * xv.x);
    atomic_add_f32(yp + 1, s * xv.y);
    atomic_add_f32(yp + 2, s * xv.z);
    atomic_add_f32(yp + 3, s * xv.w);
  }
}

// out = sigmoid(b + sum_t Tx_t @ W_t) via V_WMMA_F32_16X16X4_F32.
// One wave -> one 16x16 tile; 8 waves/block share one W column-tile staged in
// LDS (transposed so each B fragment is one aligned ds_load_b64). Staging uses
// GLOBAL_LOAD_ASYNC_TO_LDS_B32 (ASYNCcnt) when the builtin exists.
// Fragment layouts per CDNA5 ISA 7.12.2 (wave32):
//   A lane: a = { A[m][2h], A[m][2h+1] },  m = lane&15, h = lane>>4
//   B lane: b = { B[2h][n], B[2h+1][n] },  n = lane&15
//   D lane: acc[r] = D[r + 8h][n]
template <int DIN, int DOUT>
__global__ __launch_bounds__(256) void k_cheb_gemm(
    const float* __restrict__ Tx0, const float* __restrict__ Tx1,
    const float* __restrict__ Tx2, const float* __restrict__ W,
    const float* __restrict__ bias, float* __restrict__ out, int N) {
  __shared__ float sW[3 * 16 * DIN];  // sW[(t*16+n)*DIN + k] = W[t][k][ncol0+n]

  const int lane = threadIdx.x & 31;
  const int wave = threadIdx.x >> 5;
  const int mtile = blockIdx.x * 8 + wave;
  const int ncol0 = blockIdx.y * 16;
  const bool active = (mtile * 16 < N);

  // --- Stage W tile (all 256 threads; trip count exact & uniform) ---
#pragma unroll
  for (int idx = threadIdx.x; idx < 3 * 16 * DIN; idx += 256) {
    int t = idx / (16 * DIN);
    int rem = idx - t * 16 * DIN;
    int k = rem >> 4;          // consecutive threads -> consecutive n: coalesced
    int n = rem & 15;
    int dst = (t * 16 + n) * DIN + k;
    long srcoff = ((long)t * DIN + k) * DOUT + ncol0 + n;
#if HAVE_ASYNC_LDS
    __builtin_amdgcn_global_load_async_to_lds_b32(
        (g_i32*)(unsigned long long)(W + srcoff),
        (l_i32*)(unsigned long long)(unsigned int)(unsigned long long)(sW + dst),
        0, 0);
#else
    sW[dst] = W[srcoff];
#endif
  }
#if HAVE_ASYNC_LDS
#if __has_builtin(__builtin_amdgcn_s_wait_asynccnt)
  __builtin_amdgcn_s_wait_asynccnt(0);
#else
  asm volatile("s_wait_asynccnt 0x0" ::: "memory");
#endif
#endif
  __syncthreads();
  if (!active) return;  // wave-uniform: EXEC all-1s for every WMMA below

  const int h = lane >> 4;
  const int nl = lane & 15;
  const int row = mtile * 16 + nl;
  const int ncol = ncol0 + nl;

  v8f acc = {0.f, 0.f, 0.f, 0.f, 0.f, 0.f, 0.f, 0.f};
  const float* Xs[3] = {Tx0, Tx1, Tx2};
#pragma unroll
  for (int t = 0; t < 3; ++t) {
    const float* __restrict__ X = Xs[t] + (size_t)row * DIN;
    const float* sWt = &sW[(t * 16 + nl) * DIN];
#pragma unroll
    for (int kk = 0; kk < DIN; kk += 4) {
      int ka = kk + 2 * h;
      v2f a = *(const v2f*)(X + ka);          // global_load_b64, imm offset
      v2f b = *(const v2f*)(sWt + ka);        // ds_load_b64, imm offset
      acc = __builtin_amdgcn_wmma_f32_16x16x4_f32(false, a, false, b,
                                                  (short)0, acc, false, false);
    }
  }
  float bv = bias[ncol];
#pragma unroll
  for (int r2 = 0; r2 < 8; ++r2) {
    float v = acc[r2] + bv;
    out[(size_t)(mtile * 16 + r2 + 8 * h) * DOUT + ncol] =
        1.0f / (1.0f + __expf(-v));
  }
}

extern "C" void kernel_launch(void* const* d_in, const int* in_sizes, int n_in,
                              void* d_out, int out_size, void* d_ws, size_t ws_size,
                              hipStream_t stream) {
  const float* x = (const float*)d_in[0];
  const long long* ei = (const long long*)d_in[1];
  const float* ew = (const float*)d_in[2];
  const float* Ws[4] = {(const float*)d_in[3], (const float*)d_in[5],
                        (const float*)d_in[7], (const float*)d_in[9]};
  const float* bs[4] = {(const float*)d_in[4], (const float*)d_in[6],
                        (const float*)d_in[8], (const float*)d_in[10]};
  const int dims[5] = {64, 64, 64, 32, 16};
  const int N = in_sizes[0] / dims[0];
  const int E = in_sizes[2];

  auto rup = [](size_t b) { return (b + 255) & ~(size_t)255; };
  char* base = (char*)d_ws;
  size_t off = 0;
  float* dinv = (float*)(base + off); off += rup((size_t)N * 4);
  float* nrm  = (float*)(base + off); off += rup((size_t)E * 4);
  float* tx1  = (float*)(base + off); off += rup((size_t)N * 64 * 4);
  float* tx2  = (float*)(base + off); off += rup((size_t)N * 64 * 4);
  float* hA   = (float*)(base + off); off += rup((size_t)N * 64 * 4);
  float* hB   = (float*)(base + off); off += rup((size_t)N * 64 * 4);
  (void)ws_size; (void)n_in; (void)out_size;

  const int T = 256;
  auto gb = [&](long n) { return dim3((unsigned)((n + T - 1) / T)); };

  // Edge normalization (computed once, reused by all 4 layers)
  k_zero<<<gb(N), T, 0, stream>>>(dinv, N);
  k_deg<<<gb(E), T, 0, stream>>>(ei, ew, dinv, E);
  k_dinv<<<gb(N), T, 0, stream>>>(dinv, N);
  k_norm<<<gb(E), T, 0, stream>>>(ei, ew, dinv, nrm, E);

  const float* cur = x;
  float* outs[4] = {hA, hB, hA, (float*)d_out};
  for (int l = 0; l < 4; ++l) {
    int Din = dims[l], Dout = dims[l + 1];
    long nf = (long)N * Din;
    long nth = (long)E * (Din / 4);

    // Tx1 = prop(Tx0)
    k_zero<<<gb(nf), T, 0, stream>>>(tx1, nf);
    if (Din == 64) k_spmm<64><<<gb(nth), T, 0, stream>>>(ei, nrm, cur, tx1, E, 1.0f);
    else           k_spmm<32><<<gb(nth), T, 0, stream>>>(ei, nrm, cur, tx1, E, 1.0f);

    // Tx2 = 2*prop(Tx1) - Tx0  (init to -Tx0, scatter with scale 2)
    k_negcopy<<<gb(nf), T, 0, stream>>>(tx2, cur, nf);
    if (Din == 64) k_spmm<64><<<gb(nth), T, 0, stream>>>(ei, nrm, tx1, tx2, E, 2.0f);
    else           k_spmm<32><<<gb(nth), T, 0, stream>>>(ei, nrm, tx1, tx2, E, 2.0f);

    // out = sigmoid(Tx0@W0 + Tx1@W1 + Tx2@W2 + b) via fp32 WMMA
    dim3 g((unsigned)(((N + 15) / 16 + 7) / 8), (unsigned)(Dout / 16));
    if (Din == 64 && Dout == 64)
      k_cheb_gemm<64, 64><<<g, T, 0, stream>>>(cur, tx1, tx2, Ws[l], bs[l], outs[l], N);
    else if (Din == 64 && Dout == 32)
      k_cheb_gemm<64, 32><<<g, T, 0, stream>>>(cur, tx1, tx2, Ws[l], bs[l], outs[l], N);
    else
      k_cheb_gemm<32, 16><<<g, T, 0, stream>>>(cur, tx1, tx2, Ws[l], bs[l], outs[l], N);
    cur = outs[l];
  }
}